// GCN_14594298872380
// MI455X (gfx1250) — compile-verified
//
#include <hip/hip_runtime.h>

// ---------------------------------------------------------------------------
// GCN: x -> GCNConv(W1,b1) -> BN -> ReLU -> GCNConv(W2,b2)
// Memory-bound (~400MB HBM, ~3.5GB L2); GEMMs are 4.9 GFLOP total, so we use
// full-precision V_WMMA_F32_16X16X4_F32 (wave32, 16x16 C tile, 8 acc VGPRs).
// N = 100000 = 6250 * 16 exactly -> no remainder tiles.
// ---------------------------------------------------------------------------

typedef __attribute__((ext_vector_type(2))) float v2f;
typedef __attribute__((ext_vector_type(8))) float v8f;

#define IN_C  128
#define HID_C 128
#define OUT_C 64
#define BN_EPS 1e-5f

__device__ __forceinline__ void atomAddF(float* p, float v) {
    unsafeAtomicAdd(p, v);   // hardware GLOBAL_ATOMIC_ADD_F32
}

// ---------------- tiny elementwise kernels ----------------
__global__ void k_fill(float* __restrict__ p, float v, int n) {
    int i = blockIdx.x * blockDim.x + threadIdx.x;
    if (i < n) p[i] = v;
}

__global__ void k_deg_count(const int* __restrict__ col, float* __restrict__ deg, int nE) {
    int e = blockIdx.x * blockDim.x + threadIdx.x;
    if (e < nE) atomAddF(&deg[col[e]], 1.0f);
}

__global__ void k_rsqrt(float* __restrict__ d, int n) {
    int i = blockIdx.x * blockDim.x + threadIdx.x;
    if (i < n) {
        float v = d[i];
        d[i] = v > 0.0f ? __frsqrt_rn(v) : 0.0f;
    }
}

// ---------------- fp32 WMMA GEMM: Out[n x NC] = A[n x KD] @ W[KD x NC] -------
// One wave computes one 16x16 tile; 256-thread block = 8 waves.
// W staged once per block in LDS; persistent grid-stride over 16-row strips.
template <int KD, int NC>
__global__ __launch_bounds__(256) void k_gemm_wmma(const float* __restrict__ A,
                                                   const float* __restrict__ W,
                                                   float* __restrict__ Out,
                                                   int nRows) {
    __shared__ float Wl[KD * NC];
    for (int i = threadIdx.x; i < KD * NC; i += blockDim.x) Wl[i] = W[i];
    __syncthreads();

    constexpr int tilesN       = NC / 16;      // 8 (NC=128) or 4 (NC=64)
    constexpr int stripsPerBlk = 8 / tilesN;   // 1 or 2
    const int wave = threadIdx.x >> 5;
    const int lane = threadIdx.x & 31;
    const int nt   = wave % tilesN;            // N-tile handled by this wave
    const int sw   = wave / tilesN;            // strip slot within block
    const int m    = lane & 15;                // A-row / B,D-column within tile
    const int kh   = (lane >> 4) << 1;         // half-wave picks K pair 0/1 vs 2/3
    const int col0 = nt * 16;
    const int nStrips = nRows >> 4;

    for (int s = blockIdx.x * stripsPerBlk + sw; s < nStrips;
         s += gridDim.x * stripsPerBlk) {
        const float* arow = A + (size_t)(s * 16 + m) * KD;
        v8f acc = {};
#pragma unroll
        for (int kb = 0; kb < KD; kb += 4) {
            const int k0 = kb + kh;
            v2f a; a.x = arow[k0];
                   a.y = arow[k0 + 1];
            v2f b; b.x = Wl[k0 * NC + col0 + m];
                   b.y = Wl[(k0 + 1) * NC + col0 + m];
            // D = A(16x4) * B(4x16) + C   (f32, full precision)
            acc = __builtin_amdgcn_wmma_f32_16x16x4_f32(
                false, a, false, b, (short)0, acc, false, false);
        }
        const int rbase = s * 16 + ((lane >> 4) << 3);   // M = j + 8*(lane>=16)
#pragma unroll
        for (int j = 0; j < 8; ++j)
            Out[(size_t)(rbase + j) * NC + col0 + m] = acc[j];
    }
}

// ---------------- aggregation: AGG[i] = bias + dinv[i]^2 * H[i] (self loop) --
template <int C>
__global__ void k_agg_init(const float* __restrict__ H, const float* __restrict__ dinv,
                           const float* __restrict__ bias, float* __restrict__ AGG,
                           int n) {
    int idx = blockIdx.x * blockDim.x + threadIdx.x;
    int i = idx / C, c = idx % C;
    if (i < n) {
        float d = dinv[i];
        AGG[idx] = bias[c] + d * d * H[idx];
    }
}

// one wave per edge; C/32 floats per lane, vector gather + fp32 atomics
template <int C>
__global__ void k_agg_edges(const int* __restrict__ rowv, const int* __restrict__ colv,
                            const float* __restrict__ dinv, const float* __restrict__ H,
                            float* __restrict__ AGG, int nE) {
    int e = (blockIdx.x * blockDim.x + threadIdx.x) >> 5;
    if (e >= nE) return;
    const int lane = threadIdx.x & 31;
    const int src = rowv[e];
    const int dst = colv[e];
    const float norm = dinv[src] * dinv[dst];
    if (C == 128) {
        const float4 v = *(const float4*)(H + (size_t)src * 128 + lane * 4);
        float* p = AGG + (size_t)dst * 128 + lane * 4;
        atomAddF(p + 0, norm * v.x);
        atomAddF(p + 1, norm * v.y);
        atomAddF(p + 2, norm * v.z);
        atomAddF(p + 3, norm * v.w);
    } else {
        const float2 v = *(const float2*)(H + (size_t)src * 64 + lane * 2);
        float* p = AGG + (size_t)dst * 64 + lane * 2;
        atomAddF(p + 0, norm * v.x);
        atomAddF(p + 1, norm * v.y);
    }
}

// ---------------- BatchNorm: stats then apply+ReLU (in place) ----------------
__global__ void k_bn_stats(const float* __restrict__ H, float* __restrict__ stats, int n) {
    __shared__ float s1[256], s2[256];
    const int tid = threadIdx.x;
    const int c  = tid & 127;
    const int rg = tid >> 7;            // 2 row groups per block
    float sum = 0.0f, sumsq = 0.0f;
    for (int r = blockIdx.x * 2 + rg; r < n; r += gridDim.x * 2) {
        float v = H[(size_t)r * HID_C + c];
        sum += v; sumsq += v * v;
    }
    s1[tid] = sum; s2[tid] = sumsq;
    __syncthreads();
    if (tid < 128) {
        atomAddF(&stats[c],        s1[tid] + s1[tid + 128]);
        atomAddF(&stats[128 + c],  s2[tid] + s2[tid + 128]);
    }
}

__global__ void k_bn_apply(float* __restrict__ H, const float* __restrict__ stats,
                           const float* __restrict__ gamma, const float* __restrict__ beta,
                           int n) {
    int idx = blockIdx.x * blockDim.x + threadIdx.x;
    if (idx >= n * HID_C) return;
    const int c = idx & (HID_C - 1);
    const float invN = 1.0f / (float)n;
    const float mean = stats[c] * invN;
    const float var  = stats[128 + c] * invN - mean * mean;
    float v = (H[idx] - mean) * __frsqrt_rn(var + BN_EPS) * gamma[c] + beta[c];
    H[idx] = v > 0.0f ? v : 0.0f;
}

// ---------------------------------------------------------------------------
extern "C" void kernel_launch(void* const* d_in, const int* in_sizes, int n_in,
                              void* d_out, int out_size, void* d_ws, size_t ws_size,
                              hipStream_t stream) {
    const float* x     = (const float*)d_in[0];
    const int*   ei    = (const int*)  d_in[1];   // [2, E] flat: row then col
    const float* W1    = (const float*)d_in[2];
    const float* b1    = (const float*)d_in[3];
    const float* gamma = (const float*)d_in[4];
    const float* beta  = (const float*)d_in[5];
    const float* W2    = (const float*)d_in[6];
    const float* b2    = (const float*)d_in[7];
    float* out = (float*)d_out;

    const int N = in_sizes[0] / IN_C;
    const int E = in_sizes[1] / 2;
    const int* rowv = ei;
    const int* colv = ei + E;

    // workspace layout (floats), 256-elt aligned regions
    float* ws = (float*)d_ws;
    const size_t nPad = ((size_t)N + 255) & ~(size_t)255;
    float* dinv  = ws;                         // N        (deg -> rsqrt in place)
    float* H1    = dinv + nPad;                // N*128    (also reused as H3: N*64)
    float* AGG1  = H1 + nPad * HID_C;          // N*128
    float* stats = AGG1 + nPad * HID_C;        // 256
    float* H3    = H1;                         // reuse after AGG1 is built

    const int T = 256;
    auto blk = [](long long n, int t) { return (int)((n + t - 1) / t); };

    // degrees (self-loop contributes 1), then deg^-1/2
    k_fill<<<blk(N, T), T, 0, stream>>>(dinv, 1.0f, N);
    k_fill<<<1, T, 0, stream>>>(stats, 0.0f, 256);
    k_deg_count<<<blk(E, T), T, 0, stream>>>(colv, dinv, E);
    k_rsqrt<<<blk(N, T), T, 0, stream>>>(dinv, N);

    // layer 1: H1 = x @ W1 (fp32 WMMA)
    k_gemm_wmma<IN_C, HID_C><<<1024, T, 0, stream>>>(x, W1, H1, N);
    // aggregate: self loop + edges
    k_agg_init<HID_C><<<blk((long long)N * HID_C, T), T, 0, stream>>>(H1, dinv, b1, AGG1, N);
    k_agg_edges<HID_C><<<blk(E, 8), T, 0, stream>>>(rowv, colv, dinv, H1, AGG1, E);

    // batchnorm + relu (in place on AGG1)
    k_bn_stats<<<512, T, 0, stream>>>(AGG1, stats, N);
    k_bn_apply<<<blk((long long)N * HID_C, T), T, 0, stream>>>(AGG1, stats, gamma, beta, N);

    // layer 2: H3 = h @ W2 (fp32 WMMA), aggregate into d_out
    k_gemm_wmma<HID_C, OUT_C><<<1024, T, 0, stream>>>(AGG1, W2, H3, N);
    k_agg_init<OUT_C><<<blk((long long)N * OUT_C, T), T, 0, stream>>>(H3, dinv, b2, out, N);
    k_agg_edges<OUT_C><<<blk(E, 8), T, 0, stream>>>(rowv, colv, dinv, H3, out, E);
}